// GCN_15668040696438
// MI455X (gfx1250) — compile-verified
//
#include <hip/hip_runtime.h>

#define N_NODES 100000
#define N_EDGES 3200000
#define F_IN 64
#define F_OUT 128

typedef __attribute__((ext_vector_type(2))) float v2f;
typedef __attribute__((ext_vector_type(8))) float v8f;

// ---- workspace layout (bytes) ----------------------------------------------
#define A_OFF     0u
#define A_BYTES   (N_NODES * F_IN * 4u)          // 25,600,000
#define DEG_OFF   (A_OFF + A_BYTES)              // 25,600,000
#define ROW_OFF   (DEG_OFF + N_NODES * 4u)       // 26,000,000
#define CUR_OFF   (ROW_OFF + N_NODES * 4u)       // 26,400,000
#define SRCW_OFF  (CUR_OFF + N_NODES * 4u)       // 26,800,000
#define WS_NEEDED (SRCW_OFF + N_EDGES * 8u)      // 52,400,000

// ---------------------------------------------------------------------------
// Generic zero-fill (float4 granularity)
// ---------------------------------------------------------------------------
__global__ void gcn_zero_b128(float4* __restrict__ p, int n4) {
    int i = blockIdx.x * blockDim.x + threadIdx.x;
    if (i < n4) p[i] = make_float4(0.f, 0.f, 0.f, 0.f);
}

// ---------------------------------------------------------------------------
// CSR build phase 1: per-destination degree histogram (int atomics)
// ---------------------------------------------------------------------------
__global__ void __launch_bounds__(256)
gcn_count(const long long* __restrict__ edge_idx, int* __restrict__ deg) {
    const int e = blockIdx.x * blockDim.x + threadIdx.x;
    if (e < N_EDGES) atomicAdd(&deg[(int)edge_idx[e]], 1);
}

// ---------------------------------------------------------------------------
// CSR build phase 2: exclusive scan of deg -> rowStart, cursor (single block)
// Hillis-Steele over 256-chunks with running carry; ~391 iterations.
// ---------------------------------------------------------------------------
__global__ void __launch_bounds__(256)
gcn_scan(const int* __restrict__ deg, int* __restrict__ rowStart,
         int* __restrict__ cursor) {
    __shared__ int buf[256];
    __shared__ int carry_s;
    const int t = threadIdx.x;
    if (t == 0) carry_s = 0;
    __syncthreads();
    for (int base = 0; base < N_NODES; base += 256) {
        const int i = base + t;
        const int d = (i < N_NODES) ? deg[i] : 0;
        buf[t] = d;
        __syncthreads();
        #pragma unroll
        for (int off = 1; off < 256; off <<= 1) {
            const int v = (t >= off) ? buf[t - off] : 0;
            __syncthreads();
            buf[t] += v;
            __syncthreads();
        }
        const int incl = buf[t];
        const int excl = incl - d + carry_s;
        if (i < N_NODES) { rowStart[i] = excl; cursor[i] = excl; }
        __syncthreads();
        if (t == 255) carry_s += incl;   // chunk total
        __syncthreads();
    }
}

// ---------------------------------------------------------------------------
// CSR build phase 3: bucket-fill (src, weight) records per destination.
// One 8 B scatter + one int atomic per edge.
// ---------------------------------------------------------------------------
__global__ void __launch_bounds__(256)
gcn_fill(const long long* __restrict__ edge_idx,
         const float* __restrict__ edge_w,
         int* __restrict__ cursor, int2* __restrict__ srcw) {
    const int e = blockIdx.x * blockDim.x + threadIdx.x;
    if (e >= N_EDGES) return;
    const int dst = (int)edge_idx[e];
    const int src = (int)edge_idx[N_EDGES + e];
    const int pos = atomicAdd(&cursor[dst], 1);
    srcw[pos] = make_int2(src, __float_as_int(edge_w[e]));
}

// ---------------------------------------------------------------------------
// Aggregate: one wave32 per node, no float atomics.
// Lanes cooperatively load 32 edge records, broadcast each via __shfl
// (ds_bpermute), accumulate float2/lane in registers, one coalesced 256 B
// row store. Gathers hit L2 (node_embeds = 25.6 MB << 192 MB).
// ---------------------------------------------------------------------------
__global__ void __launch_bounds__(256)
gcn_aggregate(const float* __restrict__ node_embeds,
              const int* __restrict__ rowStart, const int* __restrict__ deg,
              const int2* __restrict__ srcw, float* __restrict__ A) {
    const int v    = blockIdx.x * 8 + (threadIdx.x >> 5);   // 100000 = 12500*8
    const int lane = threadIdx.x & 31;
    const int begin = rowStart[v];
    const int end   = begin + deg[v];

    float2 acc = make_float2(0.f, 0.f);
    for (int e0 = begin; e0 < end; e0 += 32) {
        const int idx = e0 + lane;
        const int2 p = (idx < end) ? srcw[idx] : make_int2(0, 0);
        const int m = min(32, end - e0);
        for (int j = 0; j < m; ++j) {
            const int   s = __shfl(p.x, j, 32);
            const float w = __int_as_float(__shfl(p.y, j, 32));
            const float2 x = ((const float2*)(node_embeds + (size_t)s * F_IN))[lane];
            acc.x += x.x * w;
            acc.y += x.y * w;
        }
    }
    ((float2*)(A + (size_t)v * F_IN))[lane] = acc;
}

// ---------------------------------------------------------------------------
// Fallback scatter (used only if ws_size can't hold the CSR): one wave/edge,
// 2 global_atomic_add_f32 per lane.
// ---------------------------------------------------------------------------
__global__ void __launch_bounds__(256)
gcn_scatter_edges(const float* __restrict__ node_embeds,
                  const long long* __restrict__ edge_idx,
                  const float* __restrict__ edge_w,
                  float* __restrict__ A) {
    const int edge = (blockIdx.x * blockDim.x + threadIdx.x) >> 5;
    const int lane = threadIdx.x & 31;
    if (edge >= N_EDGES) return;
    const long long dst = edge_idx[edge];
    const long long src = edge_idx[N_EDGES + edge];
    const float     w   = edge_w[edge];
    const float2 x = ((const float2*)(node_embeds + src * F_IN))[lane];
    float* out = A + dst * F_IN + 2 * lane;
    atomicAdd(out + 0, x.x * w);
    atomicAdd(out + 1, x.y * w);
}

// ---------------------------------------------------------------------------
// GEMM: out = relu(A @ W + bias) with V_WMMA_F32_16X16X4_F32 (exact fp32).
// Block = 8 waves; block owns 16 rows; wave w owns 16 output columns.
// ---------------------------------------------------------------------------
__global__ void __launch_bounds__(256)
gcn_gemm_bias_relu(const float* __restrict__ A,
                   const float* __restrict__ W,     // [F_IN x F_OUT]
                   const float* __restrict__ bias,  // [F_OUT]
                   float* __restrict__ out) {       // [N_NODES x F_OUT]
    __shared__ float Atile[16 * F_IN];              // 4 KB

    const int row0 = blockIdx.x * 16;
    const int tid  = threadIdx.x;

    ((float4*)Atile)[tid] = ((const float4*)(A + row0 * F_IN))[tid];
    __syncthreads();

    const int waveId = tid >> 5;
    const int lane   = tid & 31;
    const int m      = lane & 15;
    const int khalf  = lane >> 4;
    const int col0   = waveId * 16;
    const int n      = m;

    v8f c = {};
    #pragma unroll
    for (int kk = 0; kk < F_IN / 4; ++kk) {
        const int kb = kk * 4 + khalf * 2;
        v2f a, b;
        a.x = Atile[m * F_IN + kb + 0];
        a.y = Atile[m * F_IN + kb + 1];
        b.x = W[(kb + 0) * F_OUT + col0 + n];
        b.y = W[(kb + 1) * F_OUT + col0 + n];
        c = __builtin_amdgcn_wmma_f32_16x16x4_f32(
                false, a, false, b, (short)0, c, false, false);
    }

    const float bn = bias[col0 + n];
    #pragma unroll
    for (int i = 0; i < 8; ++i) {
        const int row = row0 + i + khalf * 8;
        float val = c[i] + bn;
        out[row * F_OUT + col0 + n] = val > 0.f ? val : 0.f;
    }
}

// ---------------------------------------------------------------------------
extern "C" void kernel_launch(void* const* d_in, const int* in_sizes, int n_in,
                              void* d_out, int out_size, void* d_ws, size_t ws_size,
                              hipStream_t stream) {
    const float*     node_embeds = (const float*)d_in[0];
    const long long* edge_idx    = (const long long*)d_in[1];
    const float*     edge_w      = (const float*)d_in[2];
    const float*     W           = (const float*)d_in[3];
    const float*     bias        = (const float*)d_in[4];
    float*           out         = (float*)d_out;

    char* ws = (char*)d_ws;
    float* A        = (float*)(ws + A_OFF);
    int*   deg      = (int*)  (ws + DEG_OFF);
    int*   rowStart = (int*)  (ws + ROW_OFF);
    int*   cursor   = (int*)  (ws + CUR_OFF);
    int2*  srcw     = (int2*) (ws + SRCW_OFF);

    const int edgeBlocks = N_EDGES / 256;           // 12500 exact

    if (ws_size >= (size_t)WS_NEEDED) {
        // --- CSR gather path: 6.4M int atomics instead of 204.8M f32 atomics
        gcn_zero_b128<<<(N_NODES * 4 / 16 + 255) / 256, 256, 0, stream>>>(
            (float4*)deg, N_NODES * 4 / 16);
        gcn_count<<<edgeBlocks, 256, 0, stream>>>(edge_idx, deg);
        gcn_scan<<<1, 256, 0, stream>>>(deg, rowStart, cursor);
        gcn_fill<<<edgeBlocks, 256, 0, stream>>>(edge_idx, edge_w, cursor, srcw);
        gcn_aggregate<<<N_NODES / 8, 256, 0, stream>>>(
            node_embeds, rowStart, deg, srcw, A);
    } else {
        // --- fallback: direct f32 atomic scatter
        const int n4 = N_NODES * F_IN / 4;
        gcn_zero_b128<<<(n4 + 255) / 256, 256, 0, stream>>>((float4*)A, n4);
        gcn_scatter_edges<<<(N_EDGES * 32) / 256, 256, 0, stream>>>(
            node_embeds, edge_idx, edge_w, A);
    }

    gcn_gemm_bias_relu<<<N_NODES / 16, 256, 0, stream>>>(A, W, bias, out);
}